// GraphProd2Vec_27539330302214
// MI455X (gfx1250) — compile-verified
//
#include <hip/hip_runtime.h>
#include <math.h>

// ---------------------------------------------------------------------------
// GraphProd2Vec on MI455X (gfx1250): fp32 WMMA everywhere.
//   N=8192 nodes, DIN=256, DOUT=128, E=262144 edges.
//   emb  = LN(tanh(LN(tanh(LN(x@w1+b1))@w2+b2))@w3+b3)        (3 WMMA GEMMs)
//   GAT  : fused masked-softmax aggregation, never materializing e[N][N].
// Weights are pre-transposed so B fragments are contiguous b64 loads with
// immediate offsets (compile-time strides via templates).
// ---------------------------------------------------------------------------

typedef __attribute__((ext_vector_type(2))) float v2f;
typedef __attribute__((ext_vector_type(8))) float v8f;

#define NN    8192
#define KDIM  256
#define DOUT  128
#define NEDGE 262144
#define LNEPS 1e-5f

static __device__ __forceinline__ v8f wmma4(v2f a, v2f b, v8f c) {
  // V_WMMA_F32_16X16X4_F32 : D = A(16x4) * B(4x16) + C(16x16), all f32
  return __builtin_amdgcn_wmma_f32_16x16x4_f32(false, a, false, b,
                                               (short)0, c, false, false);
}

// ---------------------------------------------------------------------------
// Adjacency bitmask: 8192 rows x 256 u32 words (bit j of row i <=> edge i~j).
// ---------------------------------------------------------------------------
__global__ void zero_mask_kernel(uint4* __restrict__ m, int nvec) {
  uint4 z = {0u, 0u, 0u, 0u};
  for (int i = blockIdx.x * blockDim.x + threadIdx.x; i < nvec;
       i += gridDim.x * blockDim.x)
    m[i] = z;
}

__global__ void build_mask_kernel(const int* __restrict__ ei,
                                  unsigned* __restrict__ mask) {
  int e = blockIdx.x * blockDim.x + threadIdx.x;
  if (e >= NEDGE) return;
  unsigned s = (unsigned)(ei[e] - 1);
  unsigned d = (unsigned)(ei[NEDGE + e] - 1);
  atomicOr(&mask[s * 256u + (d >> 5)], 1u << (d & 31u));
  atomicOr(&mask[d * 256u + (s >> 5)], 1u << (s & 31u));
}

// ---------------------------------------------------------------------------
// W[K][C] -> WT[C][K]   (K = 256 always; C = 256 or 128)
// ---------------------------------------------------------------------------
__global__ void transpose_kernel(const float* __restrict__ in,
                                 float* __restrict__ out, int cols) {
  int idx = blockIdx.x * blockDim.x + threadIdx.x;
  if (idx >= KDIM * cols) return;
  int r = idx / cols, c = idx - r * cols;           // in is [KDIM][cols]
  out[c * KDIM + r] = in[idx];
}

// ---------------------------------------------------------------------------
// Encoder layer: Y[16 rows][WD] = LN(X[16 rows][256] @ W + b) (* tanh)
// One workgroup = 16 rows; one wave per 16-column output tile (WD/16 waves).
// K = 256 fixed -> 64 x V_WMMA_F32_16X16X4_F32 per tile.
// WtT is the transposed weight [WD][256]: B frags are contiguous b64 loads.
// ---------------------------------------------------------------------------
template <int WD, bool TANH>
__global__ void __launch_bounds__(WD * 2)
encoder_kernel(const float* __restrict__ X, const float* __restrict__ WtT,
               const float* __restrict__ bias, const float* __restrict__ gamma,
               const float* __restrict__ beta, float* __restrict__ Y) {
  __shared__ float sA[16 * KDIM];                     // 16 KB input rows
  __shared__ float sSum[16], sSq[16], sMean[16], sInv[16];

  const int tid = threadIdx.x;
  const int i0  = blockIdx.x * 16;

  // Stage the 16 input rows (float4 coalesced).
  const float4* X4  = (const float4*)(X + (size_t)i0 * KDIM);
  float4*       sA4 = (float4*)sA;
  for (int q = tid; q < 16 * (KDIM / 4); q += WD * 2) sA4[q] = X4[q];
  if (tid < 16) { sSum[tid] = 0.0f; sSq[tid] = 0.0f; }
  __syncthreads();

  const int wave = tid >> 5, lane = tid & 31;
  const int m = lane & 15, hi = lane >> 4;
  const int col = wave * 16 + m;

  // Per-lane constant bases: immediate offsets from here on.
  const float* __restrict__ wp = WtT + (size_t)col * KDIM + hi * 2;
  const float* __restrict__ ap = sA + m * KDIM + hi * 2;

  v8f c = {0.f, 0.f, 0.f, 0.f, 0.f, 0.f, 0.f, 0.f};
#pragma unroll
  for (int k = 0; k < KDIM / 4; ++k) {
    v2f a = *(const v2f*)(ap + k * 4);                // A[m][kb, kb+1]   (LDS)
    v2f b = *(const v2f*)(wp + k * 4);                // WT[col][kb,kb+1] (b64)
    c = wmma4(a, b, c);
  }

  // y = c + bias ; accumulate per-row LN statistics via LDS float atomics.
  const float bcol = bias[col];
#pragma unroll
  for (int r = 0; r < 8; ++r) {
    float y = c[r] + bcol;
    c[r] = y;
    const int mr = r + hi * 8;
    atomicAdd(&sSum[mr], y);
    atomicAdd(&sSq[mr], y * y);
  }
  __syncthreads();
  if (tid < 16) {
    const float inv = 1.0f / (float)WD;
    const float mu  = sSum[tid] * inv;
    const float var = sSq[tid] * inv - mu * mu;
    sMean[tid] = mu;
    sInv[tid]  = rsqrtf(var + LNEPS);
  }
  __syncthreads();

  const float g = gamma[col], be = beta[col];
#pragma unroll
  for (int r = 0; r < 8; ++r) {
    const int mr = r + hi * 8;
    float v = (c[r] - sMean[mr]) * sInv[mr] * g + be;
    if (TANH) v = tanhf(v);
    Y[(size_t)(i0 + mr) * WD + col] = v;
  }
}

// ---------------------------------------------------------------------------
// Fused GAT: for a 16-row block, sweep all 512 column tiles.
//   S(16x16)  = emb_i @ emb_j^T                (32 fp32 WMMAs)
//   Em        = mask ? exp(S) : 0
//   den      += rowsum(Em)
//   num      += Em(16x16) @ emb_j(16x128)      (32 fp32 WMMAs)
// Finally x_gat = num / den.  4 waves, each owns every 4th column tile.
// ---------------------------------------------------------------------------
__global__ void __launch_bounds__(128)
gat_kernel(const float* __restrict__ emb, const unsigned* __restrict__ mask,
           float* __restrict__ out) {
  __shared__ float sEI[16 * DOUT];          //  8 KB  row-block emb
  __shared__ float sEJ[4][16 * DOUT];       // 32 KB  per-wave column tile
  __shared__ float sE[4][16 * 16];          //  4 KB  per-wave masked exp tile
  __shared__ float sNum[16 * DOUT];         //  8 KB  numerator accumulator
  __shared__ float sDen[16];

  const int tid = threadIdx.x;
  const int i0  = blockIdx.x * 16;

  const float4* emb4 = (const float4*)emb;
  { // stage emb rows i0..i0+15 and zero the accumulators
    const float4* E4 = emb4 + (size_t)i0 * (DOUT / 4);
    float4* sEI4  = (float4*)sEI;
    float4* sNum4 = (float4*)sNum;
    float4  z = {0.f, 0.f, 0.f, 0.f};
    for (int q = tid; q < 16 * (DOUT / 4); q += 128) { sEI4[q] = E4[q]; sNum4[q] = z; }
    if (tid < 16) sDen[tid] = 0.0f;
  }
  __syncthreads();

  const int wave = tid >> 5, lane = tid & 31;
  const int m = lane & 15, hi = lane >> 4;

  v8f zero8 = {0.f, 0.f, 0.f, 0.f, 0.f, 0.f, 0.f, 0.f};
  v8f acc[8];
#pragma unroll
  for (int t = 0; t < 8; ++t) acc[t] = zero8;
  float den[8] = {0.f, 0.f, 0.f, 0.f, 0.f, 0.f, 0.f, 0.f};

  float*  sEJw = sEJ[wave];
  float*  sEw  = sE[wave];
  float4* sEJ4 = (float4*)sEJw;

  const float* __restrict__ aI = sEI + m * DOUT + hi * 2;   // GEMM1 A base
  const float* __restrict__ bJ = sEJw + m * DOUT + hi * 2;  // GEMM1 B base
  const float* __restrict__ aE = sEw + m * 16 + hi * 2;     // GEMM2 A base

  for (int jt = wave; jt < NN / 16; jt += 4) {
    const int j0 = jt * 16;

    // Stage emb rows j0..j0+15 into this wave's private LDS region.
#pragma unroll
    for (int q = 0; q < 16; ++q) {
      const int fi = q * 32 + lane;                     // 512 float4 per tile
      sEJ4[fi] = emb4[(size_t)j0 * (DOUT / 4) + fi];
    }

    // ---- GEMM 1: S = emb_i (16x128) @ emb_j^T (128x16) ----
    v8f s = zero8;
#pragma unroll
    for (int k = 0; k < DOUT / 4; ++k) {
      v2f a = *(const v2f*)(aI + k * 4);                // A[m][kb,kb+1]
      v2f b = *(const v2f*)(bJ + k * 4);                // B[kb,kb+1][col=m]
      s = wmma4(a, b, s);
    }

    // ---- exp + mask + den, stash masked tile for re-fragmentation ----
    const unsigned shiftBase = (unsigned)(j0 & 31);     // 0 or 16
    const int      wIdx      = j0 >> 5;
#pragma unroll
    for (int r = 0; r < 8; ++r) {
      const int mr = r + hi * 8;
      const unsigned wbits = mask[(size_t)(i0 + mr) * 256u + wIdx];
      const float ev =
          ((wbits >> (shiftBase + (unsigned)m)) & 1u) ? expf(s[r]) : 0.0f;
      den[r] += ev;
      sEw[mr * 16 + m] = ev;
    }

    // ---- GEMM 2: num += Em (16x16) @ emb_j (16x128) ----
#pragma unroll
    for (int t = 0; t < 8; ++t) {
      const int n0 = t * 16;
#pragma unroll
      for (int ks = 0; ks < 4; ++ks) {
        const int kb = ks * 4 + hi * 2;
        v2f a = *(const v2f*)(aE + ks * 4);             // A[m][kb,kb+1]
        v2f b;
        b.x = sEJw[(kb)     * DOUT + n0 + m];           // B[kb  ][n0+m]
        b.y = sEJw[(kb + 1) * DOUT + n0 + m];           // B[kb+1][n0+m]
        acc[t] = wmma4(a, b, acc[t]);
      }
    }
  }

  // Cross-wave reduction via LDS float atomics.
#pragma unroll
  for (int r = 0; r < 8; ++r) {
    const int mr = r + hi * 8;
    atomicAdd(&sDen[mr], den[r]);
#pragma unroll
    for (int t = 0; t < 8; ++t)
      atomicAdd(&sNum[mr * DOUT + t * 16 + m], acc[t][r]);
  }
  __syncthreads();

  for (int q = tid; q < 16 * DOUT; q += 128) {
    const int mr = q >> 7, n = q & (DOUT - 1);
    out[(size_t)(i0 + mr) * DOUT + n] = sNum[q] / sDen[mr];
  }
}

// ---------------------------------------------------------------------------
// Launcher.  Inputs (setup_inputs order):
//  0:x 1:edge_index 2:w1 3:b1 4:g1 5:be1 6:w2 7:b2 8:g2 9:be2 10:w3 11:b3 12:g3 13:be3
// d_out = [emb (8192x128) | x_gat (8192x128)] fp32.
// d_ws  = [mask 8MB | h 8MB | w1T 256KB | w2T 256KB | w3T 128KB].
// ---------------------------------------------------------------------------
extern "C" void kernel_launch(void* const* d_in, const int* in_sizes, int n_in,
                              void* d_out, int out_size, void* d_ws,
                              size_t ws_size, hipStream_t stream) {
  const float* x   = (const float*)d_in[0];
  const int*   ei  = (const int*)d_in[1];
  const float* w1  = (const float*)d_in[2];
  const float* b1  = (const float*)d_in[3];
  const float* g1  = (const float*)d_in[4];
  const float* be1 = (const float*)d_in[5];
  const float* w2  = (const float*)d_in[6];
  const float* b2  = (const float*)d_in[7];
  const float* g2  = (const float*)d_in[8];
  const float* be2 = (const float*)d_in[9];
  const float* w3  = (const float*)d_in[10];
  const float* b3  = (const float*)d_in[11];
  const float* g3  = (const float*)d_in[12];
  const float* be3 = (const float*)d_in[13];

  float* emb  = (float*)d_out;
  float* xgat = emb + (size_t)NN * DOUT;

  char* ws = (char*)d_ws;
  unsigned* mask = (unsigned*)ws;                          //  8 MB
  float*    h    = (float*)(ws + (8u << 20));              //  8 MB hidden acts
  float*    w1T  = (float*)(ws + (16u << 20));             //  256 KB
  float*    w2T  = w1T + KDIM * KDIM;                      //  256 KB
  float*    w3T  = w2T + KDIM * KDIM;                      //  128 KB

  const int maskWords = NN * 256;                          // 2M u32
  zero_mask_kernel<<<512, 256, 0, stream>>>((uint4*)mask, maskWords / 4);
  build_mask_kernel<<<NEDGE / 256, 256, 0, stream>>>(ei, mask);

  transpose_kernel<<<(KDIM * KDIM) / 256, 256, 0, stream>>>(w1, w1T, KDIM);
  transpose_kernel<<<(KDIM * KDIM) / 256, 256, 0, stream>>>(w2, w2T, KDIM);
  transpose_kernel<<<(KDIM * DOUT) / 256, 256, 0, stream>>>(w3, w3T, DOUT);

  // Encoder: 512 row-blocks; WD/16 waves per block.
  encoder_kernel<KDIM, true ><<<NN / 16, 512, 0, stream>>>(x, w1T, b1, g1, be1, h);
  encoder_kernel<KDIM, true ><<<NN / 16, 512, 0, stream>>>(h, w2T, b2, g2, be2, h); // in-place safe
  encoder_kernel<DOUT, false><<<NN / 16, 256, 0, stream>>>(h, w3T, b3, g3, be3, emb);

  // Fused masked GAT aggregation.
  gat_kernel<<<NN / 16, 128, 0, stream>>>(emb, mask, xgat);
}